// ArcMarginProduct_56487409877363
// MI455X (gfx1250) — compile-verified
//
#include <hip/hip_runtime.h>
#include <math.h>

// ---------------- problem constants (fixed by reference) ----------------
#define N_ROWS 512
#define D_DIM  512
#define C_CLS  50000
#define SCALE_S 30.0f
#define COS_M  0.8775825618903728f
#define SIN_M  0.4794255386042030f
#define TH_C  (-0.8775825618903728f)
#define MM_C   0.2397127693021015f

#define TILE_ROWS 128         // rows of x staged in LDS per block (8 sub-tiles of 16)
#define XS_STRIDE 516         // padded row stride: 516%64==4 -> conflict-free ds reads
                              // LDS: 128*516*4 = 264192 B of the 320 KB WGP pool

typedef __attribute__((ext_vector_type(2))) float v2f;
typedef __attribute__((ext_vector_type(8))) float v8f;

// ---------------- 1 / max(||row||, eps), one wave32 per row ----------------
__global__ void row_invnorm_kernel(const float* __restrict__ src,
                                   float* __restrict__ dst, int rows) {
    int gid  = blockIdx.x * blockDim.x + threadIdx.x;
    int row  = gid >> 5;
    int lane = gid & 31;
    if (row >= rows) return;
    const float* p = src + (size_t)row * D_DIM;
    float s = 0.0f;
#pragma unroll
    for (int i = 0; i < D_DIM / 32; ++i) {
        float v = p[lane + i * 32];   // coalesced across the wave
        s += v * v;
    }
#pragma unroll
    for (int off = 16; off > 0; off >>= 1)
        s += __shfl_xor(s, off, 32);
    if (lane == 0) dst[row] = 1.0f / fmaxf(sqrtf(s), 1e-12f);
}

// ---------------- WMMA GEMM + ArcFace margin epilogue ----------------
// grid: (ceil(C/128), N/TILE_ROWS); block: 256 threads = 8 waves.
// Block stages a 128 x D x-tile in LDS (~264 KB of the 320 KB WGP LDS).
// Wave w owns 16 classes and 8 accumulators (8 row sub-tiles) -> each B
// fragment fetched from global/L2 feeds 8 WMMAs (W read only N/128 = 4x total).
__global__ void arc_gemm_kernel(const float* __restrict__ x,
                                const float* __restrict__ w,
                                const int*   __restrict__ labels,
                                const float* __restrict__ inv_x,
                                const float* __restrict__ inv_w,
                                float* __restrict__ out) {
    __shared__ __align__(16) float xs[TILE_ROWS * XS_STRIDE];
    __shared__ float lix[TILE_ROWS];
    __shared__ int   llab[TILE_ROWS];

    const int tid     = threadIdx.x;
    const int lane    = tid & 31;       // wave32 (gfx1250)
    const int wid     = tid >> 5;
    const int rowbase = blockIdx.y * TILE_ROWS;

    // cooperative load of the TILE_ROWS x D x-tile
    for (int i = tid; i < TILE_ROWS * D_DIM; i += 256) {
        int m = i >> 9;          // D_DIM == 512
        int k = i & (D_DIM - 1);
        xs[m * XS_STRIDE + k] = x[(size_t)(rowbase + m) * D_DIM + k];
    }
    if (tid < TILE_ROWS) {
        lix[tid]  = inv_x[rowbase + tid];
        llab[tid] = labels[rowbase + tid];
    }
    __syncthreads();

    const int cbase  = blockIdx.x * 128 + wid * 16;
    const int cidx   = cbase + (lane & 15);
    const int cclamp = (cidx < C_CLS) ? cidx : (C_CLS - 1);
    const float* wrow = w + (size_t)cclamp * D_DIM;

    // A frag: lane m = lane&15; VGPR0 holds K = k0 + (lane<16?0:2), VGPR1 = K+1
    // B frag: lane n = lane&15; same K split across lane halves (C-layout analogy)
    const int koff  = (lane < 16) ? 0 : 2;
    const int mlane = lane & 15;

    v8f acc[8];
#pragma unroll
    for (int t = 0; t < 8; ++t)
        acc[t] = (v8f){0.0f, 0.0f, 0.0f, 0.0f, 0.0f, 0.0f, 0.0f, 0.0f};

#pragma unroll 2
    for (int k0 = 0; k0 < D_DIM; k0 += 4) {
        v2f b = *reinterpret_cast<const v2f*>(wrow + k0 + koff);  // global_load_b64 (L2-resident W)
#pragma unroll
        for (int t = 0; t < 8; ++t) {
            v2f a = *reinterpret_cast<const v2f*>(
                        &xs[(t * 16 + mlane) * XS_STRIDE + k0 + koff]);  // ds_load_b64
            acc[t] = __builtin_amdgcn_wmma_f32_16x16x4_f32(
                         false, a, false, b, (short)0, acc[t], false, false);
        }
    }

    const float iwc = inv_w[cclamp];
#pragma unroll
    for (int t = 0; t < 8; ++t) {
#pragma unroll
        for (int r = 0; r < 8; ++r) {
            int   mrow = t * 16 + r + ((lane < 16) ? 0 : 8);  // C/D layout: VGPR r -> M=r (+8 hi lanes)
            float cosv = acc[t][r] * lix[mrow] * iwc;
            float s2   = 1.0f + 1e-7f - cosv * cosv;
            s2         = fminf(fmaxf(s2, 0.0f), 1.0f);
            float sine = sqrtf(s2);
            float phi  = cosv * COS_M - sine * SIN_M;
            phi        = (cosv > TH_C) ? phi : (cosv - MM_C);
            float val  = ((llab[mrow] == cidx) ? phi : cosv) * SCALE_S;
            if (cidx < C_CLS)
                out[(size_t)(rowbase + mrow) * C_CLS + cidx] = val;
        }
    }
}

// ---------------- single-pass online log-sum-exp per row ----------------
__global__ void row_loss_kernel(const float* __restrict__ out,
                                const int*   __restrict__ labels,
                                float* __restrict__ rowloss) {
    const int row = blockIdx.x;
    const int t   = threadIdx.x;
    const float* p = out + (size_t)row * C_CLS;

    float m = -INFINITY, s = 0.0f;
    for (int c = t; c < C_CLS; c += 256) {
        float v = p[c];
        if (v > m) { s = s * expf(m - v) + 1.0f; m = v; }
        else       { s += expf(v - m); }
    }

    __shared__ float sm[256];
    __shared__ float ss[256];
    sm[t] = m; ss[t] = s;
    __syncthreads();
    for (int off = 128; off > 0; off >>= 1) {
        if (t < off) {
            float m2 = sm[t + off], s2 = ss[t + off];
            float M  = fmaxf(sm[t], m2);
            ss[t] = ss[t] * expf(sm[t] - M) + s2 * expf(m2 - M);
            sm[t] = M;
        }
        __syncthreads();
    }
    if (t == 0) {
        int lab = labels[row];
        rowloss[row] = (sm[0] + logf(ss[0])) - p[lab];
    }
}

__global__ void loss_mean_kernel(const float* __restrict__ rowloss,
                                 float* __restrict__ loss_out) {
    __shared__ float sd[256];
    int t = threadIdx.x;
    sd[t] = rowloss[t] + rowloss[t + 256];
    __syncthreads();
    for (int off = 128; off > 0; off >>= 1) {
        if (t < off) sd[t] += sd[t + off];
        __syncthreads();
    }
    if (t == 0) loss_out[0] = sd[0] * (1.0f / (float)N_ROWS);
}

// ---------------- launcher ----------------
extern "C" void kernel_launch(void* const* d_in, const int* in_sizes, int n_in,
                              void* d_out, int out_size, void* d_ws, size_t ws_size,
                              hipStream_t stream) {
    const float* x      = (const float*)d_in[0];   // [N, D]
    const int*   labels = (const int*)  d_in[1];   // [N]
    const float* w      = (const float*)d_in[2];   // [C, D]

    float* out  = (float*)d_out;                       // [N*C] logits
    float* loss = out + (size_t)N_ROWS * C_CLS;        // + 1 scalar

    // workspace layout: inv_x[N] | inv_w[C] | rowloss[N]
    float* inv_x   = (float*)d_ws;
    float* inv_w   = inv_x + N_ROWS;
    float* rowloss = inv_w + C_CLS;

    // 1) inverse row norms (wave32 per row)
    {
        int blocks = (N_ROWS * 32 + 255) / 256;
        row_invnorm_kernel<<<blocks, 256, 0, stream>>>(x, inv_x, N_ROWS);
    }
    {
        int blocks = (C_CLS * 32 + 255) / 256;
        row_invnorm_kernel<<<blocks, 256, 0, stream>>>(w, inv_w, C_CLS);
    }

    // 2) WMMA GEMM + margin epilogue (128-row x-tile in LDS, 8 acc per wave)
    {
        dim3 grid((C_CLS + 127) / 128, N_ROWS / TILE_ROWS);
        arc_gemm_kernel<<<grid, 256, 0, stream>>>(x, w, labels, inv_x, inv_w, out);
    }

    // 3) per-row online logsumexp loss, then mean
    row_loss_kernel<<<N_ROWS, 256, 0, stream>>>(out, labels, rowloss);
    loss_mean_kernel<<<1, 256, 0, stream>>>(rowloss, loss);
}